// HighPassButterworth_30855045055072
// MI455X (gfx1250) — compile-verified
//
#include <hip/hip_runtime.h>

// ---------------------------------------------------------------------------
// Zero-phase order-4 Butterworth high-pass (filtfilt) for (32, 262144) f32.
// MI455X (gfx1250) strategy: chunked affine-scan parallelization of the IIR
// recurrence; chunk re-runs evaluated in 16-sample state-space blocks with
// V_WMMA_F32_16X16X4_F32 (Y16 = D@X16 + Cimp@Z across 16 chunk-columns).
// Intermediate forward output (~100MB) is kept in d_ws (L2-resident, 192MB).
// Requires ws_size >= ~103 MB.
// ---------------------------------------------------------------------------

#define NC    32
#define TLEN  262144
#define PADL  15
#define T3    (3 * TLEN)            // 786432
#define ELEN  (T3 + 2 * PADL)       // 786462
#define CHL   768                   // chunk length (48 blocks of 16)
#define NPF   1024                  // full chunks per channel
#define NP    1025                  // incl. 30-sample tail chunk
#define NBLK  48
#define BWLO  341                   // first chunk overlapping needed region
#define BWHI  682                   // last backward chunk in output window
#define NBWD  683                   // backward chunks computed

// workspace layout (float offsets)
#define N_Y1    ((size_t)NC * ELEN)
#define NSTATE  ((size_t)NC * NP * 4)
#define WS_FF   (N_Y1)
#define WS_SF   (WS_FF + NSTATE)
#define WS_FB   (WS_SF + NSTATE)
#define WS_SB   (WS_FB + NSTATE)
#define WS_MATS (WS_SB + NSTATE)    // 512 floats: D[256], Cimp[64], Kimp[64], M16[16], ML[16]

typedef __attribute__((ext_vector_type(2))) float v2f;
typedef __attribute__((ext_vector_type(8))) float v8f;

struct Coef { float b0, b1, b2, b3, b4, a1, a2, a3, a4; };

__device__ __forceinline__ Coef ldcoef(const float* b, const float* a) {
  Coef c;
  c.b0 = b[0]; c.b1 = b[1]; c.b2 = b[2]; c.b3 = b[3]; c.b4 = b[4];
  c.a1 = a[1]; c.a2 = a[2]; c.a3 = a[3]; c.a4 = a[4];
  return c;
}

// DF2T step, matches reference _lfilter exactly (a0 == 1).
__device__ __forceinline__ float stepf(const Coef& cf, float xv, float4& z) {
  float y = fmaf(cf.b0, xv, z.x);
  z.x = fmaf(cf.b1, xv, z.y) - cf.a1 * y;
  z.y = fmaf(cf.b2, xv, z.z) - cf.a2 * y;
  z.z = fmaf(cf.b3, xv, z.w) - cf.a3 * y;
  z.w = cf.b4 * xv - cf.a4 * y;
  return y;
}

// xpad = concat([x reversed, x, x reversed]); q in [0, 3T)
__device__ __forceinline__ float xpad_val(const float* x, int c, int q) {
  int idx;
  if (q < TLEN)          idx = TLEN - 1 - q;
  else if (q < 2 * TLEN) idx = q - TLEN;
  else                   idx = T3 - 1 - q;
  return x[(size_t)c * TLEN + idx];
}

// ext = [odd-reflect pad 15, xpad, odd-reflect pad 15]; j in [0, ELEN)
__device__ __forceinline__ float ext_val(const float* x, int c, int j) {
  int q = j - PADL;
  if (q < 0) {
    float e0 = x[(size_t)c * TLEN + (TLEN - 1)];          // xpad[0]
    return 2.0f * e0 - xpad_val(x, c, -q);                // xpad[PADL - j]
  } else if (q >= T3) {
    float eL = x[(size_t)c * TLEN + 0];                   // xpad[3T-1]
    return 2.0f * eL - xpad_val(x, c, T3 - 2 - (q - T3));
  }
  return xpad_val(x, c, q);
}

__device__ __forceinline__ void mm4(float* o, const float* A, const float* B) {
  #pragma unroll
  for (int r = 0; r < 4; ++r)
    #pragma unroll
    for (int cc = 0; cc < 4; ++cc) {
      float s = 0.0f;
      #pragma unroll
      for (int k = 0; k < 4; ++k) s = fmaf(A[r * 4 + k], B[k * 4 + cc], s);
      o[r * 4 + cc] = s;
    }
}

__device__ __forceinline__ v8f wmma4(v2f a, v2f b, v8f c) {
  // V_WMMA_F32_16X16X4_F32 : D(16x16) = A(16x4) * B(4x16) + C
  return __builtin_amdgcn_wmma_f32_16x16x4_f32(false, a, false, b, (short)0, c,
                                               false, false);
}

// ---------------------------------------------------------------------------
// K0: build filter-derived matrices (single thread; trivial work).
//  D[i][j]=h[i-j] (16x16 lower-tri impulse conv), Cimp[t][k] (16x4),
//  Kimp[r][j] (4x16), M16 = Mz^16, ML = Mz^768.
// ---------------------------------------------------------------------------
__global__ void k0_setup(const float* b, const float* a, float* mats) {
  if (threadIdx.x != 0 || blockIdx.x != 0) return;
  Coef cf = ldcoef(b, a);

  float h[16];
  { float4 z = {0, 0, 0, 0};
    for (int t = 0; t < 16; ++t) h[t] = stepf(cf, (t == 0) ? 1.0f : 0.0f, z); }
  for (int i = 0; i < 16; ++i)
    for (int j = 0; j < 16; ++j) mats[i * 16 + j] = (i >= j) ? h[i - j] : 0.0f;

  float Mz[16];
  for (int k = 0; k < 4; ++k) {
    float4 z1 = { (float)(k == 0), (float)(k == 1), (float)(k == 2), (float)(k == 3) };
    float4 zc = z1;
    stepf(cf, 0.0f, z1);                         // one step -> column k of Mz
    Mz[0 * 4 + k] = z1.x; Mz[1 * 4 + k] = z1.y;
    Mz[2 * 4 + k] = z1.z; Mz[3 * 4 + k] = z1.w;
    for (int t = 0; t < 16; ++t) mats[256 + t * 4 + k] = stepf(cf, 0.0f, zc);
  }
  for (int j = 0; j < 16; ++j) {
    float4 z = {0, 0, 0, 0};
    for (int t = 0; t < 16; ++t) stepf(cf, (t == j) ? 1.0f : 0.0f, z);
    mats[320 + 0 * 16 + j] = z.x; mats[320 + 1 * 16 + j] = z.y;
    mats[320 + 2 * 16 + j] = z.z; mats[320 + 3 * 16 + j] = z.w;
  }
  float P[16], S256[16], Tt[16];
  for (int i = 0; i < 16; ++i) P[i] = Mz[i];
  for (int i = 1; i <= 9; ++i) {                 // P = Mz^(2^i)
    mm4(Tt, P, P);
    for (int q = 0; q < 16; ++q) P[q] = Tt[q];
    if (i == 4) for (int q = 0; q < 16; ++q) mats[384 + q] = P[q];   // M16
    if (i == 8) for (int q = 0; q < 16; ++q) S256[q] = P[q];
  }
  mm4(Tt, P, S256);                              // Mz^768
  for (int q = 0; q < 16; ++q) mats[400 + q] = Tt[q];
}

// ---------------------------------------------------------------------------
// K1: forward zero-state chunk finals (one lane per (channel, chunk)).
// ---------------------------------------------------------------------------
__global__ void __launch_bounds__(256)
k1_fwd_finals(const float* x, const float* b, const float* a, float* ff) {
  int w = blockIdx.x * blockDim.x + threadIdx.x;
  if (w >= NC * NP) return;
  int c = w / NP, p = w % NP;
  Coef cf = ldcoef(b, a);
  float4 z = {0, 0, 0, 0};
  int j0 = p * CHL;
  int len = (p == NP - 1) ? (ELEN - NPF * CHL) : CHL;
  for (int t = 0; t < len; ++t) (void)stepf(cf, ext_val(x, c, j0 + t), z);
  float* o = ff + (size_t)w * 4;
  o[0] = z.x; o[1] = z.y; o[2] = z.z; o[3] = z.w;
}

// ---------------------------------------------------------------------------
// K2/K4b: per-channel affine chunk scan: z[p+1] = ML*z[p] + f[p].
// ---------------------------------------------------------------------------
__global__ void k2_fwd_scan(const float* x, const float* zi, const float* mats,
                            const float* ff, float* sf) {
  int c = threadIdx.x;
  if (c >= NC) return;
  float M[16];
  #pragma unroll
  for (int i = 0; i < 16; ++i) M[i] = mats[400 + i];
  float e0 = x[(size_t)c * TLEN + (TLEN - 1)];   // xpad[0]
  float z0 = zi[0] * e0, z1 = zi[1] * e0, z2 = zi[2] * e0, z3 = zi[3] * e0;
  for (int p = 0; p < NP; ++p) {
    size_t o = ((size_t)c * NP + p) * 4;
    sf[o] = z0; sf[o + 1] = z1; sf[o + 2] = z2; sf[o + 3] = z3;
    float f0 = ff[o], f1 = ff[o + 1], f2 = ff[o + 2], f3 = ff[o + 3];
    float n0 = M[0]  * z0 + M[1]  * z1 + M[2]  * z2 + M[3]  * z3 + f0;
    float n1 = M[4]  * z0 + M[5]  * z1 + M[6]  * z2 + M[7]  * z3 + f1;
    float n2 = M[8]  * z0 + M[9]  * z1 + M[10] * z2 + M[11] * z3 + f2;
    float n3 = M[12] * z0 + M[13] * z1 + M[14] * z2 + M[15] * z3 + f3;
    z0 = n0; z1 = n1; z2 = n2; z3 = n3;
  }
}

__global__ void k4b_bwd_scan(const float* y1, const float* zi, const float* mats,
                             const float* fb, float* sb) {
  int c = threadIdx.x;
  if (c >= NC) return;
  float M[16];
  #pragma unroll
  for (int i = 0; i < 16; ++i) M[i] = mats[400 + i];
  float u0 = y1[(size_t)c * ELEN + (ELEN - 1)];  // reversed-sequence first sample
  float z0 = zi[0] * u0, z1 = zi[1] * u0, z2 = zi[2] * u0, z3 = zi[3] * u0;
  for (int p = 0; p < NBWD; ++p) {
    size_t o = ((size_t)c * NP + p) * 4;
    sb[o] = z0; sb[o + 1] = z1; sb[o + 2] = z2; sb[o + 3] = z3;
    float f0 = fb[o], f1 = fb[o + 1], f2 = fb[o + 2], f3 = fb[o + 3];
    float n0 = M[0]  * z0 + M[1]  * z1 + M[2]  * z2 + M[3]  * z3 + f0;
    float n1 = M[4]  * z0 + M[5]  * z1 + M[6]  * z2 + M[7]  * z3 + f1;
    float n2 = M[8]  * z0 + M[9]  * z1 + M[10] * z2 + M[11] * z3 + f2;
    float n3 = M[12] * z0 + M[13] * z1 + M[14] * z2 + M[15] * z3 + f3;
    z0 = n0; z1 = n1; z2 = n2; z3 = n3;
  }
}

// ---------------------------------------------------------------------------
// K3: forward re-run with WMMA. One wave = 16 chunk-columns; 48 blocks of 16
// samples: Y = D@X + Cimp@Z via 5 chained V_WMMA_F32_16X16X4_F32.
// A(16x4) layout: lane m(=lane&15), half h(=lane>>4): {A[m][2h], A[m][2h+1]}.
// B(4x16) layout: lane n(=lane&15): {B[2h][n], B[2h+1][n]}.
// C/D: lane n, VGPR r -> Y[r + 8h][n].
// ---------------------------------------------------------------------------
__global__ void __launch_bounds__(256)
k3_fwd_rerun(const float* x, const float* mats, const float* sf, float* y1) {
  int tid = blockIdx.x * blockDim.x + threadIdx.x;
  int lane = threadIdx.x & 31;
  int W = tid >> 5;                 // global wave id (0..1023)
  int n = lane & 15, h = lane >> 4;
  int w = W * 16 + n;               // column worker
  int c = w >> 10;                  // / NPF
  int p = w & (NPF - 1);
  int pw0 = (W * 16) & (NPF - 1);   // wave-uniform chunk base
  if (pw0 + 15 < BWLO) return;      // uniform skip: y1 region not needed

  // constant operand setup
  v2f Ad[4];
  #pragma unroll
  for (int kb = 0; kb < 4; ++kb) {
    Ad[kb].x = mats[n * 16 + 4 * kb + 2 * h];
    Ad[kb].y = mats[n * 16 + 4 * kb + 2 * h + 1];
  }
  v2f Ac;
  Ac.x = mats[256 + n * 4 + 2 * h];
  Ac.y = mats[256 + n * 4 + 2 * h + 1];
  float Mr[4][2], Kr[4][8];
  #pragma unroll
  for (int r = 0; r < 4; ++r) {
    Mr[r][0] = mats[384 + r * 4 + 2 * h];
    Mr[r][1] = mats[384 + r * 4 + 2 * h + 1];
    #pragma unroll
    for (int s = 0; s < 8; ++s) {
      int t = 4 * (s >> 1) + 2 * h + (s & 1);
      Kr[r][s] = mats[320 + r * 16 + t];
    }
  }
  const float* sp = sf + ((size_t)c * NP + p) * 4;
  float za = sp[2 * h], zb = sp[2 * h + 1];

  size_t ybc = (size_t)c * ELEN;
  int base = p * CHL;
  for (int blk = 0; blk < NBLK; ++blk) {
    int j0 = base + blk * 16;
    float X[8];
    #pragma unroll
    for (int s = 0; s < 8; ++s) {
      int t = 4 * (s >> 1) + 2 * h + (s & 1);
      X[s] = ext_val(x, c, j0 + t);
    }
    v8f acc = {};
    #pragma unroll
    for (int kb = 0; kb < 4; ++kb) {
      v2f bv; bv.x = X[kb * 2]; bv.y = X[kb * 2 + 1];
      acc = wmma4(Ad[kb], bv, acc);
    }
    v2f zv; zv.x = za; zv.y = zb;
    acc = wmma4(Ac, zv, acc);

    size_t yb = ybc + (size_t)j0 + 8 * h;
    #pragma unroll
    for (int r = 0; r < 8; ++r) y1[yb + r] = acc[r];

    float part[4];
    #pragma unroll
    for (int r = 0; r < 4; ++r) {
      float sacc = Mr[r][0] * za + Mr[r][1] * zb;
      #pragma unroll
      for (int s = 0; s < 8; ++s) sacc = fmaf(Kr[r][s], X[s], sacc);
      part[r] = sacc;
    }
    #pragma unroll
    for (int r = 0; r < 4; ++r) part[r] += __shfl_xor(part[r], 16, 32);
    za = h ? part[2] : part[0];
    zb = h ? part[3] : part[1];
  }
}

// K3t: 30-sample tail chunk, scalar (one lane per channel).
__global__ void k3_tail(const float* x, const float* b, const float* a,
                        const float* sf, float* y1) {
  int c = threadIdx.x;
  if (c >= NC) return;
  Coef cf = ldcoef(b, a);
  const float* s = sf + ((size_t)c * NP + NPF) * 4;
  float4 z = {s[0], s[1], s[2], s[3]};
  for (int t = 0; t < ELEN - NPF * CHL; ++t) {
    int j = NPF * CHL + t;
    y1[(size_t)c * ELEN + j] = stepf(cf, ext_val(x, c, j), z);
  }
}

// ---------------------------------------------------------------------------
// K4a: backward zero-state chunk finals over u[t] = y1[E-1-t].
// ---------------------------------------------------------------------------
__global__ void __launch_bounds__(256)
k4a_bwd_finals(const float* y1, const float* b, const float* a, float* fb) {
  int w = blockIdx.x * blockDim.x + threadIdx.x;
  if (w >= NC * NBWD) return;
  int c = w / NBWD, p = w % NBWD;
  Coef cf = ldcoef(b, a);
  float4 z = {0, 0, 0, 0};
  size_t ub = (size_t)c * ELEN + (ELEN - 1);
  int t0 = p * CHL;
  for (int t = 0; t < CHL; ++t) (void)stepf(cf, y1[ub - (size_t)(t0 + t)], z);
  size_t o = ((size_t)c * NP + p) * 4;
  fb[o] = z.x; fb[o + 1] = z.y; fb[o + 2] = z.z; fb[o + 3] = z.w;
}

// ---------------------------------------------------------------------------
// K5: backward re-run (WMMA, same block math); only the output window
// t in [T+15, 2T+14] is stored: out[c][2T+14 - t].
// ---------------------------------------------------------------------------
__global__ void __launch_bounds__(256)
k5_bwd_rerun(const float* y1, const float* mats, const float* sb, float* out) {
  int tid = blockIdx.x * blockDim.x + threadIdx.x;
  int lane = threadIdx.x & 31;
  int W = tid >> 5;
  int n = lane & 15, h = lane >> 4;
  int w = W * 16 + n;
  int c = w >> 10;
  int p = w & (NPF - 1);
  int pw0 = (W * 16) & (NPF - 1);
  if (pw0 + 15 < BWLO || pw0 > BWHI) return;   // wave-uniform window skip

  v2f Ad[4];
  #pragma unroll
  for (int kb = 0; kb < 4; ++kb) {
    Ad[kb].x = mats[n * 16 + 4 * kb + 2 * h];
    Ad[kb].y = mats[n * 16 + 4 * kb + 2 * h + 1];
  }
  v2f Ac;
  Ac.x = mats[256 + n * 4 + 2 * h];
  Ac.y = mats[256 + n * 4 + 2 * h + 1];
  float Mr[4][2], Kr[4][8];
  #pragma unroll
  for (int r = 0; r < 4; ++r) {
    Mr[r][0] = mats[384 + r * 4 + 2 * h];
    Mr[r][1] = mats[384 + r * 4 + 2 * h + 1];
    #pragma unroll
    for (int s = 0; s < 8; ++s) {
      int t = 4 * (s >> 1) + 2 * h + (s & 1);
      Kr[r][s] = mats[320 + r * 16 + t];
    }
  }
  const float* sp = sb + ((size_t)c * NP + p) * 4;
  float za = sp[2 * h], zb = sp[2 * h + 1];

  size_t ybc = (size_t)c * ELEN;
  int base = p * CHL;
  const int wlo = TLEN + PADL;          // 262159
  const int whi = 2 * TLEN + PADL - 1;  // 524302
  for (int blk = 0; blk < NBLK; ++blk) {
    int j0 = base + blk * 16;
    float X[8];
    #pragma unroll
    for (int s = 0; s < 8; ++s) {
      int t = 4 * (s >> 1) + 2 * h + (s & 1);
      X[s] = y1[ybc + (size_t)(ELEN - 1 - (j0 + t))];
    }
    v8f acc = {};
    #pragma unroll
    for (int kb = 0; kb < 4; ++kb) {
      v2f bv; bv.x = X[kb * 2]; bv.y = X[kb * 2 + 1];
      acc = wmma4(Ad[kb], bv, acc);
    }
    v2f zv; zv.x = za; zv.y = zb;
    acc = wmma4(Ac, zv, acc);

    #pragma unroll
    for (int r = 0; r < 8; ++r) {
      int tt = j0 + 8 * h + r;
      if (tt >= wlo && tt <= whi)
        out[(size_t)c * TLEN + (size_t)(whi - tt)] = acc[r];
    }

    float part[4];
    #pragma unroll
    for (int r = 0; r < 4; ++r) {
      float sacc = Mr[r][0] * za + Mr[r][1] * zb;
      #pragma unroll
      for (int s = 0; s < 8; ++s) sacc = fmaf(Kr[r][s], X[s], sacc);
      part[r] = sacc;
    }
    #pragma unroll
    for (int r = 0; r < 4; ++r) part[r] += __shfl_xor(part[r], 16, 32);
    za = h ? part[2] : part[0];
    zb = h ? part[3] : part[1];
  }
}

extern "C" void kernel_launch(void* const* d_in, const int* in_sizes, int n_in,
                              void* d_out, int out_size, void* d_ws, size_t ws_size,
                              hipStream_t stream) {
  const float* x  = (const float*)d_in[0];
  const float* b  = (const float*)d_in[1];
  const float* a  = (const float*)d_in[2];
  const float* zi = (const float*)d_in[3];
  float* out = (float*)d_out;

  float* wsf  = (float*)d_ws;
  float* y1   = wsf;
  float* ff   = wsf + WS_FF;
  float* sf   = wsf + WS_SF;
  float* fb   = wsf + WS_FB;
  float* sb   = wsf + WS_SB;
  float* mats = wsf + WS_MATS;

  (void)in_sizes; (void)n_in; (void)out_size; (void)ws_size;

  k0_setup<<<1, 32, 0, stream>>>(b, a, mats);
  k1_fwd_finals<<<(NC * NP + 255) / 256, 256, 0, stream>>>(x, b, a, ff);
  k2_fwd_scan<<<1, 32, 0, stream>>>(x, zi, mats, ff, sf);
  k3_fwd_rerun<<<(NC * NPF) / 256, 256, 0, stream>>>(x, mats, sf, y1);
  k3_tail<<<1, 32, 0, stream>>>(x, b, a, sf, y1);
  k4a_bwd_finals<<<(NC * NBWD + 255) / 256, 256, 0, stream>>>(y1, b, a, fb);
  k4b_bwd_scan<<<1, 32, 0, stream>>>(y1, zi, mats, fb, sb);
  k5_bwd_rerun<<<(NC * NPF) / 256, 256, 0, stream>>>(y1, mats, sb, out);
}